// InstCriterion_91293824843897
// MI455X (gfx1250) — compile-verified
//
#include <hip/hip_runtime.h>
#include <hip/hip_bf16.h>
#include <math.h>

#define IGNORE_L  (-100)
#define C_CLS     20
#define LW        201      // L+1
#define NINST     64
#define NBATCH    4
#define NBLKJ     128      // partial blocks per instance in main pass (multiple of 4)
#define TPB       256

typedef float v2f __attribute__((ext_vector_type(2)));
typedef float v8f __attribute__((ext_vector_type(8)));

// ---------------------------------------------------------------- helpers
__device__ __forceinline__ float blkred(float v, float* red) {
    int t = threadIdx.x;
    red[t] = v;
    __syncthreads();
    for (int s = TPB / 2; s > 0; s >>= 1) {
        if (t < s) red[t] += red[t + s];
        __syncthreads();
    }
    float r = red[0];
    __syncthreads();
    return r;
}

// ------------------------------------------------ K1: per-point statistics
// scal: [0]=nll_sum [1]=nll_cnt [2]=dist_sum [3]=dir_sum [4]=valid_cnt
__global__ void k_point_stats(const float* __restrict__ sem,
                              const float* __restrict__ ptoff,
                              const float* __restrict__ iinfo,
                              const float* __restrict__ coords,
                              const int*   __restrict__ slab,
                              const int*   __restrict__ ilab,
                              const int*   __restrict__ boffs,
                              int N, float* scal, float* HF) {
    __shared__ float lhist[NBATCH * LW];
    __shared__ float red[TPB];
    for (int x = threadIdx.x; x < NBATCH * LW; x += blockDim.x) lhist[x] = 0.f;
    __syncthreads();

    float nll_s = 0.f, cnt_s = 0.f, dist_s = 0.f, dir_s = 0.f, vcnt = 0.f;

    for (int i = blockIdx.x * blockDim.x + threadIdx.x; i < N;
         i += gridDim.x * blockDim.x) {
        // --- semantic NLL (log-softmax) ---
        const float* row = sem + (size_t)i * C_CLS;
        float m = row[0];
        #pragma unroll
        for (int c = 1; c < C_CLS; c++) m = fmaxf(m, row[c]);
        float se = 0.f;
        #pragma unroll
        for (int c = 0; c < C_CLS; c++) se += expf(row[c] - m);
        int  sl = slab[i];
        bool sv = (sl != IGNORE_L);
        int  lab = sv ? sl : 0;
        float lse = m + logf(se);
        if (sv) { nll_s += (lse - row[lab]); cnt_s += 1.f; }

        // --- offset losses ---
        float gx = iinfo[(size_t)i * 9 + 0] - coords[(size_t)i * 3 + 0];
        float gy = iinfo[(size_t)i * 9 + 1] - coords[(size_t)i * 3 + 1];
        float gz = iinfo[(size_t)i * 9 + 2] - coords[(size_t)i * 3 + 2];
        float px = ptoff[(size_t)i * 3 + 0];
        float py = ptoff[(size_t)i * 3 + 1];
        float pz = ptoff[(size_t)i * 3 + 2];
        int   il  = ilab[i];
        float val = (il != IGNORE_L) ? 1.f : 0.f;
        dist_s += (fabsf(px - gx) + fabsf(py - gy) + fabsf(pz - gz)) * val;
        float gn = sqrtf(gx * gx + gy * gy + gz * gz) + 1e-8f;
        float pn = sqrtf(px * px + py * py + pz * pz) + 1e-8f;
        dir_s  += -((gx * px + gy * py + gz * pz) / (gn * pn)) * val;
        vcnt   += val;

        // --- hist_full: batch x labels_id ---
        int lid = (il == IGNORE_L) ? 0 : (il + 1);
        int lo = 0, hi = NBATCH + 1;                 // searchsorted right
        while (lo < hi) { int mid = (lo + hi) >> 1; if (boffs[mid] <= i) lo = mid + 1; else hi = mid; }
        int b = min(max(lo - 1, 0), NBATCH - 1);
        atomicAdd(&lhist[b * LW + lid], 1.0f);
    }

    float acc[5] = {nll_s, cnt_s, dist_s, dir_s, vcnt};
    #pragma unroll
    for (int a = 0; a < 5; a++) {
        float s = blkred(acc[a], red);
        if (threadIdx.x == 0) atomicAdd(&scal[a], s);
    }
    __syncthreads();
    for (int x = threadIdx.x; x < NBATCH * LW; x += blockDim.x)
        if (lhist[x] != 0.f) atomicAdd(&HF[x], lhist[x]);
}

// ------------------------------------------------ K2: proposal histogram
__global__ void k_prop_hist(const int* __restrict__ pidx,
                            const int* __restrict__ poffs,
                            const int* __restrict__ ilab,
                            int T, float* HI) {
    int t = blockIdx.x * blockDim.x + threadIdx.x;
    if (t >= T) return;
    int lo = 0, hi = NINST + 1;                       // searchsorted right on 65 offsets
    while (lo < hi) { int mid = (lo + hi) >> 1; if (poffs[mid] <= t) lo = mid + 1; else hi = mid; }
    int seg = min(max(lo - 1, 0), NINST - 1);
    int p   = pidx[2 * t + 1];
    int il  = ilab[p];
    int lid = (il == IGNORE_L) ? 0 : (il + 1);
    atomicAdd(&HI[seg * LW + lid], 1.0f);
}

// ------------------------------------------------ K3: masked-point histogram
__global__ void k_mask_hist(const int* __restrict__ oidx,
                            const int* __restrict__ ilab,
                            const int* __restrict__ bids,
                            int M, float* HM) {
    int j = blockIdx.x * blockDim.x + threadIdx.x;
    if (j >= M) return;
    int oi  = oidx[j];
    int il  = ilab[oi];
    int lid = (il == IGNORE_L) ? 0 : (il + 1);
    atomicAdd(&HM[bids[j] * LW + lid], 1.0f);
}

// ------------------------------------------------ K4: mode/skip/cover/pos
__global__ void k_mode(const float* __restrict__ HI,
                       const float* __restrict__ HM,
                       const float* __restrict__ HF,
                       const int* __restrict__ ibids,
                       const int* __restrict__ ipsl,
                       int* MODE, int* SKIP, int* POS) {
    int i = threadIdx.x;
    if (i < NINST) {
        const float* h = HI + i * LW;
        int best = 0; float bv = h[0];
        for (int l = 1; l < LW; l++) { float v = h[l]; if (v > bv) { bv = v; best = l; } }
        int sk = (best == 0) || (ipsl[i] <= 3);
        int b  = ibids[i];
        float num = HM[b * LW + best];
        float den = HF[b * LW + best];
        float cover = (den > 0.f) ? (num / den) : 0.f;
        MODE[i] = best;
        SKIP[i] = sk;
        POS[i]  = (!sk) && (cover > 0.3f);
    }
}

// ------------------------------------------------ K5: main INSTxM pass
// P layout: P[((a*NINST)+i)*NBLKJ + blk], a in {0:wsum,1:inter,2:up,3:ug,4:bce}
__global__ void k_mask_main(const float* __restrict__ mlog,
                            const int* __restrict__ oidx,
                            const int* __restrict__ slab,
                            const int* __restrict__ ilab,
                            const int* __restrict__ bids,
                            const int* __restrict__ ibids,
                            const int* __restrict__ ipsl,
                            const int* __restrict__ MODE,
                            const int* __restrict__ SKIP,
                            const int* __restrict__ POS,
                            int M, float* P) {
    __shared__ float red[TPB];
    int i     = blockIdx.y;
    int mode  = MODE[i];
    int sk    = SKIP[i];
    int ps    = POS[i];
    int bi    = ibids[i];
    int labi  = ipsl[i];
    const float* lrow = mlog + (size_t)i * M;

    int chunk = (M + NBLKJ - 1) / NBLKJ;
    int j0 = blockIdx.x * chunk;
    int j1 = min(j0 + chunk, M);

    float wsum = 0.f, inter = 0.f, up = 0.f, ug = 0.f, bces = 0.f;
    for (int j = j0 + threadIdx.x; j < j1; j += TPB) {
        int oi  = oidx[j];
        int il  = ilab[oi];
        int lid = (il == IGNORE_L) ? 0 : (il + 1);
        int sm  = slab[oi];
        float w  = (bids[j] == bi && (sk || sm == labi)) ? 1.f : 0.f;
        float gt = (ps && lid == mode) ? 1.f : 0.f;
        float x  = lrow[j];
        float pred = 1.f / (1.f + expf(-x));
        float sp   = fmaxf(x, 0.f) + log1pf(expf(-fabsf(x)));   // stable softplus
        wsum  += w;
        inter += w * pred * gt;
        up    += w * pred * pred;
        ug    += w * gt;
        bces  += w * (sp - gt * x);
    }

    float acc[5] = {wsum, inter, up, ug, bces};
    #pragma unroll
    for (int a = 0; a < 5; a++) {
        float s = blkred(acc[a], red);
        if (threadIdx.x == 0)
            P[((size_t)a * NINST + i) * NBLKJ + blockIdx.x] = s;
    }
}

// ------------------------------------------------ K6: WMMA reduction + final loss
// One wave32, EXEC all ones.  V_WMMA_F32_16X16X4_F32 with B==1 performs an
// exact f32 row-reduction: D[m,n] = sum_k A[m,k]*1 + C[m,n].
// A layout (32-bit A 16x4): lanes 0-15 -> M=lane, K={0,1}; lanes 16-31 -> M=lane-16, K={2,3}.
// C/D layout: VGPR r: lanes 0-15 -> row r, lanes 16-31 -> row r+8.
__global__ void k_final(const float* __restrict__ P,
                        const float* __restrict__ scal,
                        const int* __restrict__ SKIP,
                        const int* __restrict__ epoch,
                        float* out) {
    __shared__ float sums[5 * NINST];
    int lane = threadIdx.x;
    int m    = lane & 15;
    int koff = (lane < 16) ? 0 : 2;
    const v2f onesB = {1.f, 1.f};

    for (int a = 0; a < 5; a++) {
        for (int g = 0; g < NINST / 16; g++) {
            v8f acc = {};
            const float* base = P + ((size_t)a * NINST + g * 16 + m) * NBLKJ + koff;
            #pragma unroll 4
            for (int t = 0; t < NBLKJ / 4; t++) {
                v2f A = { base[4 * t], base[4 * t + 1] };
                acc = __builtin_amdgcn_wmma_f32_16x16x4_f32(
                          false, A, false, onesB, (short)0, acc, false, false);
            }
            if (lane == 0) {
                #pragma unroll
                for (int r = 0; r < 8; r++) sums[a * NINST + g * 16 + r] = acc[r];
            }
            if (lane == 16) {
                #pragma unroll
                for (int r = 0; r < 8; r++) sums[a * NINST + g * 16 + 8 + r] = acc[r];
            }
        }
    }
    __syncthreads();

    if (lane == 0) {
        float sem   = scal[0] / fmaxf(scal[1], 1.f);
        float vs    = scal[4] + 1e-6f;
        float loss  = sem + scal[2] / vs + scal[3] / vs;
        if (epoch[0] > 128) {
            float wtot = 0.f, btot = 0.f, dsum = 0.f, dcnt = 0.f;
            for (int i = 0; i < NINST; i++) {
                float wsum = sums[0 * NINST + i];
                float intr = sums[1 * NINST + i];
                float up   = sums[2 * NINST + i];
                float ug   = sums[3 * NINST + i];
                float bce  = sums[4 * NINST + i];
                wtot += wsum;
                btot += bce;
                if (!SKIP[i] && wsum > 0.f) {
                    dsum += 1.f - 2.f * intr / (up + ug + 1e-5f);
                    dcnt += 1.f;
                }
            }
            loss += btot / (wtot + 1e-6f) + dsum / (dcnt + 1e-6f);
        }
        out[0] = loss;
    }
}

// ---------------------------------------------------------------- launcher
extern "C" void kernel_launch(void* const* d_in, const int* in_sizes, int n_in,
                              void* d_out, int out_size, void* d_ws, size_t ws_size,
                              hipStream_t stream) {
    const float* sem    = (const float*)d_in[0];
    const float* ptoff  = (const float*)d_in[1];
    const float* iinfo  = (const float*)d_in[2];
    const float* coords = (const float*)d_in[3];
    const float* mlog   = (const float*)d_in[4];
    const int*   slab   = (const int*)d_in[5];
    const int*   ilab   = (const int*)d_in[6];
    const int*   oidx   = (const int*)d_in[7];
    const int*   pidx   = (const int*)d_in[8];
    const int*   poffs  = (const int*)d_in[9];
    const int*   ibids  = (const int*)d_in[10];
    const int*   bids   = (const int*)d_in[11];
    const int*   boffs  = (const int*)d_in[12];
    const int*   ipsl   = (const int*)d_in[13];
    const int*   epoch  = (const int*)d_in[14];

    int N = in_sizes[0] / C_CLS;
    int M = in_sizes[4] / NINST;
    int T = in_sizes[8] / 2;

    float* W    = (float*)d_ws;
    float* scal = W;                      // 8
    float* HF   = W + 8;                  // NBATCH*LW
    float* HI   = HF + NBATCH * LW;       // NINST*LW
    float* HM   = HI + NINST * LW;        // NBATCH*LW
    float* P    = HM + NBATCH * LW;       // 5*NINST*NBLKJ
    int*   MODE = (int*)(P + 5 * NINST * NBLKJ);
    int*   SKIP = MODE + NINST;
    int*   POS  = SKIP + NINST;
    size_t zbytes = (size_t)((char*)(POS + NINST) - (char*)d_ws);
    (void)hipMemsetAsync(d_ws, 0, zbytes, stream);

    k_point_stats<<<512, TPB, 0, stream>>>(sem, ptoff, iinfo, coords, slab, ilab,
                                           boffs, N, scal, HF);
    k_prop_hist<<<(T + TPB - 1) / TPB, TPB, 0, stream>>>(pidx, poffs, ilab, T, HI);
    k_mask_hist<<<(M + TPB - 1) / TPB, TPB, 0, stream>>>(oidx, ilab, bids, M, HM);
    k_mode<<<1, 64, 0, stream>>>(HI, HM, HF, ibids, ipsl, MODE, SKIP, POS);
    dim3 g5(NBLKJ, NINST);
    k_mask_main<<<g5, TPB, 0, stream>>>(mlog, oidx, slab, ilab, bids, ibids, ipsl,
                                        MODE, SKIP, POS, M, P);
    k_final<<<1, 32, 0, stream>>>(P, scal, SKIP, epoch, (float*)d_out);
}